// BiLSTM_2559800508939
// MI455X (gfx1250) — compile-verified
//
#include <hip/hip_runtime.h>
#include <hip/hip_bf16.h>
#include <math.h>

typedef __attribute__((ext_vector_type(2))) float v2f;
typedef __attribute__((ext_vector_type(8))) float v8f;

#define T_LEN 256
#define I_DIM 46
#define I_PAD 48
#define H_DIM 64
#define G_DIM 256   /* 4*H */
#define B_DIM 2048
#define NC 8
#define HPAD (H_DIM + 2)   /* keeps 8B alignment for paired A-frag loads */

// CDNA5 hardware TANH transcendental. The trailing v_nop satisfies the ISA
// TRANS-op hazard rule (one op must separate a trans op from the use of its result),
// which the compiler cannot see through inline asm.
__device__ __forceinline__ float hw_tanh(float x) {
    float r;
    asm volatile("v_tanh_f32 %0, %1\n\tv_nop" : "=v"(r) : "v"(x));
    return r;
}
__device__ __forceinline__ float sigmoid_t(float z) {
    // exactly 1/(1+exp(-z)) via the tanh identity; one TRANS instruction
    return 0.5f + 0.5f * hw_tanh(0.5f * z);
}

__device__ __forceinline__ void load_xfrag(v2f (&xa)[12], const float* __restrict__ xt,
                                           int hi) {
#pragma unroll
    for (int kc = 0; kc < 12; ++kc) {
        const int kk = kc * 4 + hi * 2;          // this lane's K pair
        if (kk < I_DIM) { xa[kc].x = xt[kk]; xa[kc].y = xt[kk + 1]; }
        else            { xa[kc].x = 0.f;    xa[kc].y = 0.f; }   // pad rows 46,47
    }
}

// -------------------------------------------------------------------------
// Forward LSTM scan. Block = 16-row batch tile, 4 waves; wave w owns hidden
// columns [16w,16w+16) of ALL four gates so c stays in a v8f register and
// gate fusion is lane-local.
//   - Whh B-fragments: weight-stationary in 128 VGPRs (loaded once).
//   - Wih B-fragments: pre-paired float2 in LDS -> single aligned b64 loads.
//   - x  A-fragments: register double-buffered; loads for t+1 issue right
//     after the last use at step t, hidden under 64 WMMAs + activations.
// Matrix math: V_WMMA_F32_16X16X4_F32 (exact f32 = reference math).
// -------------------------------------------------------------------------
__global__ __launch_bounds__(128) void lstm_fwd_kernel(
    const float* __restrict__ x,     // (B, T, I)
    const float* __restrict__ Wih,   // (4H, I)
    const float* __restrict__ Whh,   // (4H, H)
    const float* __restrict__ bih,   // (4H)
    const float* __restrict__ bhh,   // (4H)
    float* __restrict__ hf_out)      // (B, H)
{
    __shared__ v2f   pwih[I_PAD / 2][G_DIM];   // packed Wih^T K-pairs (48 KB)
    __shared__ float hbuf[2][16][HPAD];        // double-buffered h (~8.4 KB)

    const int tid    = threadIdx.x;
    const int wave   = tid >> 5;
    const int lane   = tid & 31;
    const int lane15 = lane & 15;
    const int hi     = lane >> 4;            // 0: lanes 0-15, 1: lanes 16-31
    const int b0     = blockIdx.x * 16;
    const int ncol   = wave * 16 + lane15;   // hidden unit owned by this lane

    // Stage Wih^T into LDS pre-paired: pwih[k2][g] = (WihT[2k2][g], WihT[2k2+1][g])
    for (int idx = tid; idx < (I_PAD / 2) * G_DIM; idx += 128) {
        int k2 = idx / G_DIM, g = idx % G_DIM;
        int i0 = 2 * k2;
        v2f v;
        v.x = (i0     < I_DIM) ? Wih[g * I_DIM + i0    ] : 0.0f;
        v.y = (i0 + 1 < I_DIM) ? Wih[g * I_DIM + i0 + 1] : 0.0f;
        pwih[k2][g] = v;
    }
    for (int idx = tid; idx < 16 * HPAD; idx += 128)
        hbuf[0][idx / HPAD][idx % HPAD] = 0.0f;   // h0 = 0

    // Weight-stationary Whh B-fragments: whf[kc][g] = (WhhT[kk][col], WhhT[kk+1][col]),
    // kk = 4*kc + 2*hi, col = g*64 + ncol.
    v2f whf[16][4];
#pragma unroll
    for (int kc = 0; kc < 16; ++kc) {
        const int kk = kc * 4 + hi * 2;
#pragma unroll
        for (int g = 0; g < 4; ++g) {
            const float* wp = Whh + (size_t)(g * H_DIM + ncol) * H_DIM + kk;
            whf[kc][g].x = wp[0];
            whf[kc][g].y = wp[1];
        }
    }

    // Per-lane gate biases (column-constant within each 16x16 D tile).
    float bias[4];
#pragma unroll
    for (int g = 0; g < 4; ++g)
        bias[g] = bih[g * H_DIM + ncol] + bhh[g * H_DIM + ncol];

    __syncthreads();

    v8f c = {0.f, 0.f, 0.f, 0.f, 0.f, 0.f, 0.f, 0.f};
    const float* xrow = x + (size_t)(b0 + lane15) * T_LEN * I_DIM;

    v2f xa[12];
    load_xfrag(xa, xrow, hi);                     // prime t = 0

    for (int t = 0; t < T_LEN; ++t) {
        const int cur = t & 1, nxt = cur ^ 1;

        v8f acc[4];
#pragma unroll
        for (int g = 0; g < 4; ++g) {
#pragma unroll
            for (int r = 0; r < 8; ++r) acc[g][r] = bias[g];
        }

        // ---- z += x_t @ Wih^T  (K = 48, zero-padded; A from registers) ----
#pragma unroll
        for (int kc = 0; kc < 12; ++kc) {
#pragma unroll
            for (int g = 0; g < 4; ++g) {
                v2f bf = pwih[kc * 2 + hi][g * H_DIM + ncol];   // one ds_load_b64
                acc[g] = __builtin_amdgcn_wmma_f32_16x16x4_f32(
                    false, xa[kc], false, bf, (short)0, acc[g], false, false);
            }
        }

        // ---- prefetch x_{t+1} fragments; latency covered by the section below ----
        const int tn = (t + 1 < T_LEN) ? t + 1 : t;
        load_xfrag(xa, xrow + tn * I_DIM, hi);

        // ---- z += h_{t-1} @ Whh^T  (K = 64, B from registers) ----
#pragma unroll
        for (int kc = 0; kc < 16; ++kc) {
            const int kk = kc * 4 + hi * 2;
            v2f a = *(const v2f*)&hbuf[cur][lane15][kk];        // aligned b64
#pragma unroll
            for (int g = 0; g < 4; ++g) {
                acc[g] = __builtin_amdgcn_wmma_f32_16x16x4_f32(
                    false, a, false, whf[kc][g], (short)0, acc[g], false, false);
            }
        }

        // ---- gate nonlinearities + state update (lane-local, HW TANH) ----
#pragma unroll
        for (int r = 0; r < 8; ++r) {
            float ig = sigmoid_t(acc[0][r]);
            float fg = sigmoid_t(acc[1][r]);
            float gg = hw_tanh(acc[2][r]);
            float og = sigmoid_t(acc[3][r]);
            float cn = fg * c[r] + ig * gg;
            c[r] = cn;
            hbuf[nxt][r + hi * 8][ncol] = og * hw_tanh(cn);
        }
        __syncthreads();
    }

    // Final forward hidden lives in hbuf[0] (written at t = T-1, T even).
    for (int idx = tid; idx < 16 * H_DIM; idx += 128) {
        int m = idx / H_DIM, j = idx % H_DIM;
        hf_out[(size_t)(b0 + m) * H_DIM + j] = hbuf[0][m][j];
    }
}

// -------------------------------------------------------------------------
// Tail: backward direction contributes only ONE cell step (h0=c0=0, f-gate
// multiplies zero) on x[:,T-1,:], then FC + softmax. One thread per batch.
// -------------------------------------------------------------------------
__global__ __launch_bounds__(256) void lstm_tail_kernel(
    const float* __restrict__ x,       // (B, T, I)
    const float* __restrict__ Wih_b,   // (4H, I)
    const float* __restrict__ bih_b,   // (4H)
    const float* __restrict__ bhh_b,   // (4H)
    const float* __restrict__ fcW,     // (NC, 2H)
    const float* __restrict__ fcb,     // (NC)
    const float* __restrict__ hf,      // (B, H) from forward kernel
    float* __restrict__ out)           // (B, NC)
{
    int b = blockIdx.x * blockDim.x + threadIdx.x;
    if (b >= B_DIM) return;

    float xv[I_DIM];
    const float* xp = x + ((size_t)b * T_LEN + (T_LEN - 1)) * I_DIM;
#pragma unroll
    for (int i = 0; i < I_DIM; ++i) xv[i] = xp[i];

    float logits[NC];
#pragma unroll
    for (int k = 0; k < NC; ++k) logits[k] = fcb[k];

    // forward-hidden half of the FC
    for (int j = 0; j < H_DIM; ++j) {
        float h = hf[(size_t)b * H_DIM + j];
#pragma unroll
        for (int k = 0; k < NC; ++k) logits[k] += h * fcW[k * (2 * H_DIM) + j];
    }

    // backward single-cell + its half of the FC (gate order i,f,g,o; f dead)
    for (int j = 0; j < H_DIM; ++j) {
        float zi = bih_b[j]             + bhh_b[j];
        float zg = bih_b[2 * H_DIM + j] + bhh_b[2 * H_DIM + j];
        float zo = bih_b[3 * H_DIM + j] + bhh_b[3 * H_DIM + j];
        const float* wi = Wih_b + (size_t)j * I_DIM;
        const float* wg = Wih_b + (size_t)(2 * H_DIM + j) * I_DIM;
        const float* wo = Wih_b + (size_t)(3 * H_DIM + j) * I_DIM;
        for (int i = 0; i < I_DIM; ++i) {
            zi += xv[i] * wi[i];
            zg += xv[i] * wg[i];
            zo += xv[i] * wo[i];
        }
        float ig = sigmoid_t(zi);
        float gg = hw_tanh(zg);
        float og = sigmoid_t(zo);
        float hb = og * hw_tanh(ig * gg);   // c = i*g since c0 = 0
#pragma unroll
        for (int k = 0; k < NC; ++k)
            logits[k] += hb * fcW[k * (2 * H_DIM) + H_DIM + j];
    }

    // softmax
    float mx = logits[0];
#pragma unroll
    for (int k = 1; k < NC; ++k) mx = fmaxf(mx, logits[k]);
    float s = 0.f;
#pragma unroll
    for (int k = 0; k < NC; ++k) { logits[k] = __expf(logits[k] - mx); s += logits[k]; }
    float inv = 1.f / s;
#pragma unroll
    for (int k = 0; k < NC; ++k) out[(size_t)b * NC + k] = logits[k] * inv;
}

extern "C" void kernel_launch(void* const* d_in, const int* in_sizes, int n_in,
                              void* d_out, int out_size, void* d_ws, size_t ws_size,
                              hipStream_t stream) {
    const float* x     = (const float*)d_in[0];
    const float* Wih_f = (const float*)d_in[1];
    const float* Whh_f = (const float*)d_in[2];
    const float* bih_f = (const float*)d_in[3];
    const float* bhh_f = (const float*)d_in[4];
    const float* Wih_b = (const float*)d_in[5];
    // d_in[6] (Whh_b) is mathematically dead: backward output uses h0 = 0.
    const float* bih_b = (const float*)d_in[7];
    const float* bhh_b = (const float*)d_in[8];
    const float* fcW   = (const float*)d_in[9];
    const float* fcb   = (const float*)d_in[10];

    float* hf = (float*)d_ws;   // (B, H) = 512 KB scratch

    lstm_fwd_kernel<<<B_DIM / 16, 128, 0, stream>>>(x, Wih_f, Whh_f, bih_f, bhh_f, hf);
    lstm_tail_kernel<<<B_DIM / 256, 256, 0, stream>>>(x, Wih_b, bih_b, bhh_b,
                                                      fcW, fcb, hf, (float*)d_out);
}